// ChamferDistance_4733053960750
// MI455X (gfx1250) — compile-verified
//
#include <hip/hip_runtime.h>

typedef __attribute__((ext_vector_type(2))) float v2f;
typedef __attribute__((ext_vector_type(8))) float v8f;

#define WAVES_PER_BLOCK 8

__global__ void chamfer_init_ws(float* ws) {
    ws[0] = 0.0f;
    ws[1] = 0.0f;
}

// One wave owns a 16-row tile of P (rows n0..n0+15) and scans all columns of Q
// in 16-wide tiles. Each 16x16 tile of (sq2[m] - 2*dot(p1[n], p2[m])) is one
// v_wmma_f32_16x16x4_f32 with a loop-invariant zero C operand (hoisted by
// LICM), the sq2 term fused into B's K=4 slot, and the sq1[n] row term applied
// once in the epilogue (it is constant over the min_m reduction).
__global__ __launch_bounds__(256) void chamfer_rowmin_wmma(
    const float* __restrict__ P, const float* __restrict__ Q,
    int N, int M, int tilesPerBatch, int numTiles,
    float* __restrict__ sumOut)
{
    const int lane = threadIdx.x & 31;
    const int wave = threadIdx.x >> 5;
    const int tile = blockIdx.x * WAVES_PER_BLOCK + wave;
    if (tile >= numTiles) return;   // wave-uniform branch

    const int b  = tile / tilesPerBatch;
    const int n0 = (tile - b * tilesPerBatch) * 16;
    const int l  = lane & 15;        // position within half-wave
    const bool hi = (lane >= 16);    // K half selector (K=0,1 vs K=2,3)

    const float* __restrict__ Pb = P + (size_t)b * 3 * (size_t)N;
    const float* __restrict__ Qb = Q + (size_t)b * 3 * (size_t)M;

    // ---- A fragment: row = n0 + l ----
    // lanes 0-15 : VGPR0=K0, VGPR1=K1  -> (-2x, -2y)
    // lanes 16-31: VGPR0=K2, VGPR1=K3  -> (-2z, 1.0)
    v2f afrag;
    float sq1;   // kept for the epilogue: lane l holds sq1[n0 + l]
    {
        const float* p = Pb + 3 * (n0 + l);
        const float px = p[0], py = p[1], pz = p[2];
        sq1 = fmaf(px, px, fmaf(py, py, pz * pz));
        afrag.x = hi ? (-2.0f * pz) : (-2.0f * px);
        afrag.y = hi ? 1.0f         : (-2.0f * py);
    }

    // Loop-invariant zero accumulator (no per-tile broadcast).
    v8f czero;
#pragma unroll
    for (int i = 0; i < 8; ++i) czero[i] = 0.0f;

    v8f rmin;
#pragma unroll
    for (int i = 0; i < 8; ++i) rmin[i] = 3.402823466e+38f;

    for (int m0 = 0; m0 < M; m0 += 16) {
        // ---- B fragment: col = m0 + l ----
        // lanes 0-15 : VGPR0=K0 row (x2), VGPR1=K1 row (y2)
        // lanes 16-31: VGPR0=K2 row (z2), VGPR1=K3 row (sq2)
        const float* q = Qb + 3 * (m0 + l);
        const float qx = q[0], qy = q[1], qz = q[2];
        const float sq2 = fmaf(qx, qx, fmaf(qy, qy, qz * qz));

        v2f bfrag;
        bfrag.x = hi ? qz  : qx;
        bfrag.y = hi ? sq2 : qy;

        // D[n][m] = sq2[m] - 2 * dot(p1[n], p2[m])
        v8f d = __builtin_amdgcn_wmma_f32_16x16x4_f32(
            /*neg_a=*/false, afrag, /*neg_b=*/false, bfrag,
            /*c_mod=*/(short)0, czero, /*reuse_a=*/false, /*reuse_b=*/false);

#pragma unroll
        for (int i = 0; i < 8; ++i) rmin[i] = fminf(rmin[i], d[i]);
    }

    // Sum of sq1 over each lane's group of 8 rows:
    // after xor{1,2,4}, lane 0 holds sum(sq1[n0+0..7]),
    // lane 24 holds sum(sq1[n0+8..15]).
    float g = sq1;
    g += __shfl_xor(g, 1, 32);
    g += __shfl_xor(g, 2, 32);
    g += __shfl_xor(g, 4, 32);

    // Reduce row mins across the 16 lanes of each half-wave.
    // half 0 lanes -> rows n0+0..7, half 1 lanes -> rows n0+8..15.
#pragma unroll
    for (int i = 0; i < 8; ++i) {
        float v = rmin[i];
        v = fminf(v, __shfl_xor(v, 1, 32));
        v = fminf(v, __shfl_xor(v, 2, 32));
        v = fminf(v, __shfl_xor(v, 4, 32));
        v = fminf(v, __shfl_xor(v, 8, 32));
        rmin[i] = v;
    }

    // lane 0: rows n0+0..7 (rmin from half 0, g = sum sq1[n0+0..7])
    // lane 24: rows n0+8..15 (rmin from half 1, g = sum sq1[n0+8..15])
    if (lane == 0 || lane == 24) {
        float s = g;
#pragma unroll
        for (int i = 0; i < 8; ++i) s += rmin[i];
        atomicAdd(sumOut, s);
    }
}

__global__ void chamfer_finalize(float* out, const float* ws, float inv1, float inv2) {
    out[0] = ws[0] * inv1 + ws[1] * inv2;
}

extern "C" void kernel_launch(void* const* d_in, const int* in_sizes, int n_in,
                              void* d_out, int out_size, void* d_ws, size_t ws_size,
                              hipStream_t stream) {
    const float* pc1 = (const float*)d_in[0];
    const float* pc2 = (const float*)d_in[1];
    float* out = (float*)d_out;
    float* ws  = (float*)d_ws;

    const int B = 2, D = 3;
    const int N = in_sizes[0] / (B * D);   // 8192
    const int M = in_sizes[1] / (B * D);   // 8192

    chamfer_init_ws<<<1, 1, 0, stream>>>(ws);

    const int tiles1 = B * (N / 16);
    const int tiles2 = B * (M / 16);
    const int blocks1 = (tiles1 + WAVES_PER_BLOCK - 1) / WAVES_PER_BLOCK;
    const int blocks2 = (tiles2 + WAVES_PER_BLOCK - 1) / WAVES_PER_BLOCK;

    // Pass 1: dist1 = min over pc2 columns for each pc1 row.
    chamfer_rowmin_wmma<<<blocks1, 256, 0, stream>>>(pc1, pc2, N, M, N / 16, tiles1, ws + 0);
    // Pass 2 (symmetric): dist2 = min over pc1 columns for each pc2 row.
    chamfer_rowmin_wmma<<<blocks2, 256, 0, stream>>>(pc2, pc1, M, N, M / 16, tiles2, ws + 1);

    chamfer_finalize<<<1, 1, 0, stream>>>(out, ws, 1.0f / (float)(B * N), 1.0f / (float)(B * M));
}